// Convolution_78005196030567
// MI455X (gfx1250) — compile-verified
//
#include <hip/hip_runtime.h>

typedef float v2f __attribute__((ext_vector_type(2)));
typedef float v8f __attribute__((ext_vector_type(8)));

#define Zb   2
#define Nn   256
#define CIN  32
#define COUT 32
#define Hh   64
#define LDP  260   // padded K-stride: multiple of 4 => conflict-free float2 LDS frags

// out[z,a,i] = sum_{h,j} G[h,j] * W2[h, i*32+j] + sum_j b2[i*32+j]*fsum[z,j]
// G[h,j]     = sum_b relu(P_b[h] + b1[h] - P_a[h]) * f[z,b,j]
// P[z,n,h]   = g[z,n,:] . W1[:,h]
__launch_bounds__(256)
__global__ void pointconv_kernel(const float* __restrict__ F,   // [Z,N,CIN]
                                 const float* __restrict__ Gm,  // [Z,N,3]
                                 const float* __restrict__ W1,  // [3,H]
                                 const float* __restrict__ B1,  // [H]
                                 const float* __restrict__ W2,  // [H, COUT*CIN]
                                 const float* __restrict__ B2,  // [COUT*CIN]
                                 float* __restrict__ out)       // [Z,N,COUT]
{
    extern __shared__ float smem[];
    float* hT   = smem;              // [64][LDP]  h transposed: hT[h][b]
    float* fT   = hT + Hh * LDP;     // [32][LDP]  f transposed: fT[j][b]
    float* gL   = fT + CIN * LDP;    // [256][3]   geometry rows for this z
    float* w1L  = gL + Nn * 3;       // [3][64]
    float* cL   = w1L + 3 * Hh;      // [64]       b1[h] - P_a[h]
    float* fsum = cL + Hh;           // [32]       sum_b f[z,b,j]
    float* Gs   = fsum + CIN;        // [64][32]   G result
    float* red  = Gs + Hh * CIN;     // [8][32]    reduction scratch

    const int tid = threadIdx.x;
    const int z   = blockIdx.x >> 8;
    const int a   = blockIdx.x & 255;

    // ---- stage transposed f, geometry, W1 into LDS ----
    const float* Fz = F + z * Nn * CIN;
    for (int idx = tid; idx < Nn * CIN; idx += 256) {
        int b = idx >> 5, j = idx & 31;
        fT[j * LDP + b] = Fz[idx];                 // coalesced global read
    }
    const float* Gz = Gm + z * Nn * 3;
    for (int idx = tid; idx < Nn * 3; idx += 256) gL[idx] = Gz[idx];
    if (tid < 3 * Hh) w1L[tid] = W1[tid];
    __syncthreads();

    // c[h] = b1[h] - P_a[h]   (MLP1 is affine in geometry: rel.W1 = P_b - P_a)
    if (tid < Hh) {
        float pa = gL[a*3+0]*w1L[tid] + gL[a*3+1]*w1L[Hh+tid] + gL[a*3+2]*w1L[2*Hh+tid];
        cL[tid] = B1[tid] - pa;
    }
    if (tid >= 64 && tid < 96) {                   // fsum[j] = sum_b f[z,b,j]
        int j = tid - 64;
        float s = 0.f;
        for (int b = 0; b < Nn; ++b) s += fT[j * LDP + b];
        fsum[j] = s;
    }
    __syncthreads();

    // hT[h][b] = relu(P_b[h] + c[h])
    for (int idx = tid; idx < Nn * Hh; idx += 256) {
        int hh = idx & 63, b = idx >> 6;
        float v = gL[b*3+0]*w1L[hh] + gL[b*3+1]*w1L[Hh+hh] + gL[b*3+2]*w1L[2*Hh+hh] + cL[hh];
        hT[hh * LDP + b] = fmaxf(v, 0.f);
    }
    __syncthreads();

    // ---- G[h][j] = sum_b hT[h][b] * fT[j][b] via V_WMMA_F32_16X16X4_F32 ----
    // 8 waves <-> 8 tiles (4 M-tiles of h x 2 N-tiles of j), K = 256 in steps of 4.
    // A frag (16x4 f32): lane<16 holds A[M=lane, K..K+1], lane>=16 holds K+2..K+3.
    // B frag (4x16 f32): lane<16 holds B[K..K+1, N=lane], lane>=16 holds K+2..K+3.
    const int wave = tid >> 5, lane = tid & 31;
    const int mt = wave >> 1, nt = wave & 1;
    const int row  = mt * 16 + (lane & 15);        // A row (h index)
    const int col  = nt * 16 + (lane & 15);        // B col (j index)
    const int koff = (lane >> 4) << 1;             // K sub-offset per half-wave
    const float* ap = hT + row * LDP + koff;
    const float* bp = fT + col * LDP + koff;

    v8f acc0 = {0.f,0.f,0.f,0.f,0.f,0.f,0.f,0.f};
    v8f acc1 = {0.f,0.f,0.f,0.f,0.f,0.f,0.f,0.f};
    for (int kk = 0; kk < Nn; kk += 8) {           // 2 accumulators break RAW chain
        v2f a0 = *(const v2f*)(ap + kk);
        v2f b0 = *(const v2f*)(bp + kk);
        acc0 = __builtin_amdgcn_wmma_f32_16x16x4_f32(false, a0, false, b0,
                                                     (short)0, acc0, false, false);
        v2f a1 = *(const v2f*)(ap + kk + 4);
        v2f b1v = *(const v2f*)(bp + kk + 4);
        acc1 = __builtin_amdgcn_wmma_f32_16x16x4_f32(false, a1, false, b1v,
                                                     (short)0, acc1, false, false);
    }
    // D layout: VGPR r -> row M0+r (lanes 0-15) / M0+8+r (lanes 16-31), N = lane&15
    const int rbase = mt * 16 + ((lane >> 4) << 3);
    #pragma unroll
    for (int r = 0; r < 8; ++r)
        Gs[(rbase + r) * CIN + col] = acc0[r] + acc1[r];
    __syncthreads();

    // ---- out[i] = sum_{h,j} Gs[h][j]*W2[h, i*32+j]  + sum_j b2[i*32+j]*fsum[j] ----
    {
        int i = tid & 31, seg = tid >> 5;          // 8 h-segments x 32 outputs
        float p = 0.f;
        for (int hh = seg * 8; hh < seg * 8 + 8; ++hh) {
            const float* w2r = W2 + hh * (COUT * CIN) + i * CIN;
            const float* gr  = Gs + hh * CIN;
            #pragma unroll
            for (int j = 0; j < CIN; ++j) p += gr[j] * w2r[j];
        }
        red[seg * 32 + i] = p;
    }
    __syncthreads();
    if (tid < COUT) {
        float s = 0.f;
        #pragma unroll
        for (int r = 0; r < 8; ++r) s += red[r * 32 + tid];
        float bias = 0.f;
        const float* b2r = B2 + tid * CIN;
        #pragma unroll
        for (int j = 0; j < CIN; ++j) bias += b2r[j] * fsum[j];
        out[(z * Nn + a) * COUT + tid] = s + bias;
    }
}

extern "C" void kernel_launch(void* const* d_in, const int* in_sizes, int n_in,
                              void* d_out, int out_size, void* d_ws, size_t ws_size,
                              hipStream_t stream) {
    (void)in_sizes; (void)n_in; (void)out_size; (void)d_ws; (void)ws_size;
    const float* F  = (const float*)d_in[0];   // features [Z,N,CIN]
    const float* Gm = (const float*)d_in[1];   // geometry [Z,N,3]
    const float* W1 = (const float*)d_in[2];   // [3,H]
    const float* B1 = (const float*)d_in[3];   // [H]
    const float* W2 = (const float*)d_in[4];   // [H, COUT*CIN]
    const float* B2 = (const float*)d_in[5];   // [COUT*CIN]
    float* out = (float*)d_out;

    const size_t shmem =
        (size_t)((Hh + CIN) * LDP + Nn * 3 + 3 * Hh + Hh + CIN + Hh * CIN + 8 * 32)
        * sizeof(float);                        // ~111 KB, legal on CDNA5 (320 KB/WG)

    pointconv_kernel<<<dim3(Zb * Nn), dim3(256), shmem, stream>>>(
        F, Gm, W1, B1, W2, B2, out);
}